// SlaterDetSampler_ordered_13649406067469
// MI455X (gfx1250) — compile-verified
//
#include <hip/hip_runtime.h>
#include <hip/hip_bf16.h>
#include <math.h>

// D = 96 sites, N = 48 particles.
// out layout: probs[48*96] | prob_sample[1] | occ_vec[96]  (all f32)

typedef __attribute__((ext_vector_type(2))) float v2f;
typedef __attribute__((ext_vector_type(8))) float v8f;

#define DD 96
#define NN 48

// ---------------------------------------------------------------------------
// Kernel 1: G = I - P P^T  via V_WMMA_F32_16X16X4_F32.
// Grid: 36 blocks (6x6 tiles of 16x16), block = 1 wave32. EXEC all ones.
// A[m,k] = P[rt+m, k]   (16x4 f32 fragment: lane%16 = M; VGPR0: K=2*half, VGPR1: K=2*half+1)
// B[k,n] = P[ct+n, k]   (4x16 f32 fragment, symmetric layout)
// ---------------------------------------------------------------------------
__global__ __launch_bounds__(32) void g_gemm_kernel(const float* __restrict__ eig,
                                                    float* __restrict__ G) {
    const int tile = blockIdx.x;          // 0..35
    const int rt = (tile / 6) * 16;
    const int ct = (tile % 6) * 16;
    const int lane = threadIdx.x;         // 0..31
    const int half = lane >> 4;           // 0 or 1
    const int idx  = lane & 15;

    v8f acc = {};
    const float* arow = eig + (size_t)(rt + idx) * DD;   // P[r, k] = eig[r*96 + k], k < 48
    const float* brow = eig + (size_t)(ct + idx) * DD;

    for (int k0 = 0; k0 < NN; k0 += 4) {
        v2f a, b;
        a.x = arow[k0 + 2 * half + 0];
        a.y = arow[k0 + 2 * half + 1];
        b.x = brow[k0 + 2 * half + 0];
        b.y = brow[k0 + 2 * half + 1];
        acc = __builtin_amdgcn_wmma_f32_16x16x4_f32(
            /*neg_a=*/false, a, /*neg_b=*/false, b,
            /*c_mod=*/(short)0, acc, /*reuse_a=*/false, /*reuse_b=*/false);
    }

    const int col = ct + idx;
#pragma unroll
    for (int v = 0; v < 8; ++v) {
        const int row = rt + v + 8 * half;   // C/D layout: VGPR v, lanes 0-15 -> M=v, 16-31 -> M=v+8
        const float diag = (row == col) ? 1.0f : 0.0f;
        G[(size_t)row * DD + col] = diag - acc[v];
    }
}

// ---------------------------------------------------------------------------
// Kernel 2: per-step-k unpivoted LU of Mbase = G - diag(occ_before[k]) in LDS,
// collecting pivots; leading minors d_m = prod(piv[0..m-1]);
// probs[k,i] = -(d_{i+1} - d_i) / d_{xmin},  masked to [xmin, xmax) & |p|>1e-15.
// Only columns < xmax are needed (minors beyond xmax are masked to 0).
// Grid: 48 blocks (one per particle step), 128 threads.
// ---------------------------------------------------------------------------
__global__ __launch_bounds__(128) void lu_probs_kernel(const float* __restrict__ G,
                                                       const int* __restrict__ pos,
                                                       float* __restrict__ out) {
    __shared__ float M[DD][DD + 1];    // +1 pad: spread across 64 banks
    __shared__ float piv[DD];
    __shared__ float d[DD + 1];

    const int k = blockIdx.x;          // 0..47
    const int tid = threadIdx.x;

    // Load G into LDS.
    for (int e = tid; e < DD * DD; e += blockDim.x) {
        M[e / DD][e % DD] = G[e];
    }
    __syncthreads();

    // Subtract occupancy of the k already-placed particles (distinct sites).
    if (tid < k) {
        const int p = pos[tid];
        M[p][p] -= 1.0f;
    }
    const int xmin = (k == 0) ? 0 : (pos[k - 1] + 1);
    const int xmax = DD - NN + k + 1;  // <= 96
    __syncthreads();

    // Partial right-looking LU restricted to the leading xmax block.
    for (int j = 0; j < xmax; ++j) {
        if (tid == 0) piv[j] = M[j][j];
        __syncthreads();
        if (j < xmax - 1) {            // uniform condition
            const float inv = 1.0f / piv[j];
            const int i = j + 1 + tid; // one row per thread (<= 95 rows)
            if (i < xmax) {
                const float L = M[i][j] * inv;
                for (int c = j + 1; c < xmax; ++c) {
                    M[i][c] -= L * M[j][c];
                }
            }
            __syncthreads();
        }
    }

    // Prefix products -> leading principal minors.
    if (tid == 0) {
        d[0] = 1.0f;
        for (int m = 1; m <= xmax; ++m) d[m] = d[m - 1] * piv[m - 1];
    }
    __syncthreads();

    if (tid < DD) {
        const int i = tid;
        float o = 0.0f;
        if (i >= xmin && i < xmax) {
            const float p = -(d[i + 1] - d[i]) / d[xmin];
            o = (fabsf(p) > 1e-15f) ? p : 0.0f;
        }
        out[(size_t)k * DD + i] = o;
    }
}

// ---------------------------------------------------------------------------
// Kernel 3: prob_sample = prod_k probs[k, positions[k]]; occ_vec one-hot sum.
// ---------------------------------------------------------------------------
__global__ __launch_bounds__(96) void finalize_kernel(const int* __restrict__ pos,
                                                      const float* __restrict__ probs,
                                                      float* __restrict__ out) {
    const int t = threadIdx.x;         // 0..95
    float occ = 0.0f;
    for (int j = 0; j < NN; ++j) {
        if (pos[j] == t) occ = 1.0f;
    }
    out[NN * DD + 1 + t] = occ;
    if (t == 0) {
        float prod = 1.0f;
        for (int kk = 0; kk < NN; ++kk) {
            prod *= probs[(size_t)kk * DD + pos[kk]];
        }
        out[NN * DD] = prod;
    }
}

extern "C" void kernel_launch(void* const* d_in, const int* in_sizes, int n_in,
                              void* d_out, int out_size, void* d_ws, size_t ws_size,
                              hipStream_t stream) {
    const float* eig = (const float*)d_in[0];   // eigfunc: 96x96 f32, row-major
    const int*   pos = (const int*)d_in[1];     // positions: 48 int32, sorted
    float* out = (float*)d_out;                 // 4705 f32
    float* G   = (float*)d_ws;                  // 96*96 f32 scratch (36 KB)

    g_gemm_kernel<<<36, 32, 0, stream>>>(eig, G);
    lu_probs_kernel<<<48, 128, 0, stream>>>(G, pos, out);
    finalize_kernel<<<1, 96, 0, stream>>>(pos, out, out);
}